// GlobalAtt_53755810677324
// MI455X (gfx1250) — compile-verified
//
#include <hip/hip_runtime.h>
#include <hip/hip_bf16.h>

typedef float v2f __attribute__((ext_vector_type(2)));
typedef float v8f __attribute__((ext_vector_type(8)));

#define DFEAT 128
#define GA_EPS 1e-16f
#define CHUNK 256  // nodes per wave in the pooling kernel

__device__ __forceinline__ void atomicMaxFloat(float* addr, float v) {
    // order-preserving int trick; segmax initialized to -inf (0xFF800000)
    if (v >= 0.0f) atomicMax(reinterpret_cast<int*>(addr), __float_as_int(v));
    else           atomicMin(reinterpret_cast<unsigned int*>(addr), __float_as_uint(v));
}

// ---------------- K0: init out tile, segmax, denom ----------------
__global__ void k_init(float* __restrict__ outp, float* __restrict__ segmax,
                       float* __restrict__ denom, int SD, int S) {
    int i = blockIdx.x * blockDim.x + threadIdx.x;
    if (i < SD) outp[i] = 0.0f;
    if (i < S) {
        segmax[i] = __int_as_float(0xFF800000);  // -inf
        denom[i]  = 0.0f;
    }
}

// ---------------- K1: gate = x @ W + b ; per-segment max ----------------
// one wave (32 lanes) per row; float4 coalesced loads; shfl reduction
__global__ void k_gate(const float* __restrict__ x, const int* __restrict__ batch,
                       const float* __restrict__ W, const float* __restrict__ b,
                       float* __restrict__ gate_out, float* __restrict__ segmax,
                       int N) {
    __shared__ float sW[DFEAT];
    for (int i = threadIdx.x; i < DFEAT; i += blockDim.x) sW[i] = W[i];
    __syncthreads();

    const int lane = threadIdx.x & 31;
    const int wid  = (blockIdx.x * blockDim.x + threadIdx.x) >> 5;
    const int nw   = (gridDim.x * blockDim.x) >> 5;
    const float bias = b[0];

    const float4* x4 = reinterpret_cast<const float4*>(x);
    const float4  w4 = reinterpret_cast<const float4*>(sW)[lane];

    for (int row = wid; row < N; row += nw) {
        float4 xv = x4[(size_t)row * (DFEAT / 4) + lane];
        float p = xv.x * w4.x + xv.y * w4.y + xv.z * w4.z + xv.w * w4.w;
        #pragma unroll
        for (int o = 16; o > 0; o >>= 1) p += __shfl_xor(p, o, 32);
        if (lane == 0) {
            float gt = p + bias;
            gate_out[row] = gt;
            atomicMaxFloat(&segmax[batch[row]], gt);
        }
    }
}

// ---------------- K2: g = exp(gate - max); denom = segsum(g) ----------------
__global__ void k_denom(const float* __restrict__ gate_in, const int* __restrict__ batch,
                        const float* __restrict__ segmax, float* __restrict__ g_out,
                        float* __restrict__ denom, int N, int S) {
    extern __shared__ float sden[];
    for (int i = threadIdx.x; i < S; i += blockDim.x) sden[i] = 0.0f;
    __syncthreads();

    int i = blockIdx.x * blockDim.x + threadIdx.x;
    if (i < N) {
        int s   = batch[i];
        float g = expf(gate_in[i] - segmax[s]);
        g_out[i] = g;
        atomicAdd(&sden[s], g);          // LDS atomic (ds_add_f32)
    }
    __syncthreads();
    for (int s = threadIdx.x; s < S; s += blockDim.x) {
        float d = sden[s];
        if (d != 0.0f) atomicAdd(&denom[s], d);
    }
}

// ---------------- K3: WMMA scatter-pooling + gate_sm write ----------------
// wave owns CHUNK sorted nodes; A(16 seg x 4 nodes) one-hot-scaled gate_sm,
// B(4 nodes x 16 cols) x-tile; 8 col-tiles accumulate 16x128 per wave.
__global__ void k_pool_wmma(const float* __restrict__ x, const int* __restrict__ batch,
                            const float* __restrict__ g_in, const float* __restrict__ denom,
                            float* __restrict__ outp, float* __restrict__ gsm_out,
                            int N, int S, int nChunks) {
    const int lane  = threadIdx.x & 31;
    const int wid   = (blockIdx.x * (blockDim.x >> 5)) + (threadIdx.x >> 5);
    if (wid >= nChunks) return;

    const int half = lane >> 4;    // 0: lanes 0-15, 1: lanes 16-31
    const int lm   = lane & 15;    // A row (M) / B,D column (N)
    const int kk   = half * 2;     // K slot base for this lane-half

    const int r0      = wid * CHUNK;
    const int rend    = min(r0 + CHUNK, N);
    const int segbase = batch[r0];

    v8f acc[8];
    #pragma unroll
    for (int t = 0; t < 8; ++t) acc[t] = (v8f)0.0f;

    for (int k0 = r0; k0 < rend; k0 += 4) {
        __builtin_prefetch(x + (size_t)min(k0 + 4, N - 1) * DFEAT + lane * 4, 0, 0);

        int   sg[4]; float gs[4];
        #pragma unroll
        for (int j = 0; j < 4; ++j) {
            int i = k0 + j;
            bool valid = i < rend;
            int  s  = valid ? batch[i] : segbase;
            float g = valid ? g_in[i]  : 0.0f;
            sg[j] = s;
            gs[j] = g / (denom[s] + GA_EPS);
            if (!valid) gs[j] = 0.0f;
        }
        // write normalized gate_sm (lanes 0..3 cover the 4 nodes)
        if (lane < 4 && (k0 + lane) < rend) gsm_out[k0 + lane] = gs[lane];

        // A matrix: lane-half holds K = kk, kk+1 for row M = lm
        int off0 = sg[kk]     - segbase;
        int off1 = sg[kk + 1] - segbase;
        v2f A;
        A.x = (off0 == lm) ? gs[kk]     : 0.0f;   // lm in [0,15] => implies window hit
        A.y = (off1 == lm) ? gs[kk + 1] : 0.0f;

        // B matrix: VGPR0 = row K=kk, VGPR1 = row K=kk+1, column = lm
        const float* xr0 = x + (size_t)min(k0 + kk,     N - 1) * DFEAT;
        const float* xr1 = x + (size_t)min(k0 + kk + 1, N - 1) * DFEAT;
        #pragma unroll
        for (int t = 0; t < 8; ++t) {
            v2f B;
            B.x = xr0[t * 16 + lm];
            B.y = xr1[t * 16 + lm];
            acc[t] = __builtin_amdgcn_wmma_f32_16x16x4_f32(
                false, A, false, B, (short)0, acc[t], false, false);
        }

        // rare fallback: node outside the 16-segment window (A contribution was 0)
        #pragma unroll
        for (int j = 0; j < 4; ++j) {
            int off = sg[j] - segbase;
            if ((off < 0 || off > 15) && (k0 + j) < rend) {
                const float* xr = x + (size_t)(k0 + j) * DFEAT;
                float gv = gs[j];
                #pragma unroll
                for (int c = 0; c < 4; ++c) {
                    int col = lane * 4 + c;
                    atomicAdd(&outp[(size_t)sg[j] * DFEAT + col], gv * xr[col]);
                }
            }
        }
    }

    // epilogue: D layout — VGPR v, lane-half => row M = v + 8*half, col N = lm
    #pragma unroll
    for (int t = 0; t < 8; ++t) {
        #pragma unroll
        for (int v = 0; v < 8; ++v) {
            float val = acc[t][v];
            if (val != 0.0f) {
                int row = segbase + v + half * 8;
                if (row < S) atomicAdd(&outp[(size_t)row * DFEAT + t * 16 + lm], val);
            }
        }
    }
}

extern "C" void kernel_launch(void* const* d_in, const int* in_sizes, int n_in,
                              void* d_out, int out_size, void* d_ws, size_t ws_size,
                              hipStream_t stream) {
    const float* x     = (const float*)d_in[0];
    const int*   batch = (const int*)d_in[1];
    const float* W     = (const float*)d_in[3];
    const float* b     = (const float*)d_in[4];

    const int N  = in_sizes[0] / DFEAT;          // nodes
    const int S  = (out_size - N) / DFEAT;       // segments
    const int SD = S * DFEAT;

    float* outp   = (float*)d_out;               // [S, D]
    float* gate   = outp + SD;                   // [N] — raw gate, later gate_sm
    float* gbuf   = (float*)d_ws;                // [N] — exp values
    float* segmax = gbuf + N;                    // [S]
    float* denom  = segmax + S;                  // [S]

    // K0: init
    {
        int n = SD;  // SD >= S always (D=128)
        k_init<<<(n + 255) / 256, 256, 0, stream>>>(outp, segmax, denom, SD, S);
    }
    // K1: gate matvec + segment max  (raw gate parked in the gate_sm output slot)
    k_gate<<<2048, 256, 0, stream>>>(x, batch, W, b, gate, segmax, N);
    // K2: exp + denominator
    k_denom<<<(N + 255) / 256, 256, (size_t)S * sizeof(float), stream>>>(
        gate, batch, segmax, gbuf, denom, N, S);
    // K3: WMMA pooling + normalized gate_sm
    {
        int nChunks = (N + CHUNK - 1) / CHUNK;
        int wavesPerBlock = 256 / 32;
        int blocks = (nChunks + wavesPerBlock - 1) / wavesPerBlock;
        k_pool_wmma<<<blocks, 256, 0, stream>>>(x, batch, gbuf, denom,
                                                outp, gate, N, S, nChunks);
    }
}